// RoIPool_28587302322329
// MI455X (gfx1250) — compile-verified
//
#include <hip/hip_runtime.h>
#include <hip/hip_bf16.h>
#include <math.h>

// Problem constants (from reference)
#define B_    2
#define C_    256
#define H_    50
#define W_    76
#define NROIS 128
#define SCALE 0.0625f
#define LDSF  8192    // floats per LDS buffer (32 KB); worst tile = 76*50 = 3800 floats

typedef unsigned int       u32;
typedef unsigned long long u64;
typedef u32 __attribute__((ext_vector_type(4))) u32x4;
typedef int __attribute__((ext_vector_type(4))) i32x4;
typedef int __attribute__((ext_vector_type(8))) i32x8;

__device__ __forceinline__ int rfl(int v) { return __builtin_amdgcn_readfirstlane(v); }

// Issue one TDM load: 3D tile (tw x th x cn) of f32 from global (strides W_, H_*W_)
// into LDS at byte offset lds_off, packed contiguously (x fastest, then y, then c).
__device__ __forceinline__ void tdm_issue(const float* gbase, u32 lds_off,
                                          int tw, int th, int cn) {
  u64 ga = (u64)(uintptr_t)gbase;
  const int HW = H_ * W_;

  u32x4 g0;
  g0[0] = (u32)rfl(1);                                   // count=1, user descriptor
  g0[1] = (u32)rfl((int)lds_off);                        // lds_addr (bytes)
  g0[2] = (u32)rfl((int)(u32)ga);                        // global_addr[31:0]
  g0[3] = (u32)rfl((int)(((u32)(ga >> 32) & 0x01FFFFFFu) // global_addr[56:32]
                        | 0x80000000u));                 // type = 2 ("image")

  i32x8 g1;
  g1[0] = rfl(2 << 16);                                  // data_size=4B; no multicast/pad/iterate
  g1[1] = rfl((tw & 0xFFFF) << 16);                      // tensor_dim0[15:0] = tw
  g1[2] = rfl((th & 0xFFFF) << 16);                      // tensor_dim0[31:16]=0; tensor_dim1[15:0]=th
  g1[3] = rfl((tw & 0xFFFF) << 16);                      // tensor_dim1[31:16]=0; tile_dim0 = tw
  g1[4] = rfl((th & 0xFFFF) | ((cn & 0xFFFF) << 16));    // tile_dim1 = th; tile_dim2 = cn
  g1[5] = rfl(W_);                                       // tensor_dim0_stride[31:0] = W
  g1[6] = rfl((HW & 0xFFFF) << 16);                      // stride0[47:32]=0; stride1[15:0] = H*W
  g1[7] = rfl((HW >> 16) & 0xFFFF);                      // stride1[47:16]

  i32x4 g2;
  g2[0] = rfl(cn);                                       // tensor_dim2 = cn
  g2[1] = rfl(1);                                        // tensor_dim3 (unused)
  g2[2] = 0;                                             // tensor_dim2_stride lo (unused)
  g2[3] = 0;                                             // stride hi + tile_dim3 = 0
  i32x4 g3 = {0, 0, 0, 0};
  i32x8 gx = {0, 0, 0, 0, 0, 0, 0, 0};                   // 6-arg toolchain extra operand

  __builtin_amdgcn_tensor_load_to_lds(g0, g1, g2, g3, gx, 0);
}

__global__ __launch_bounds__(256)
void roipool_tdm_kernel(const float* __restrict__ feat,
                        const float* __restrict__ rois,
                        float* __restrict__ out) {
  __shared__ float lds[2][LDSF];

  const int tid = threadIdx.x;
  const int roi = blockIdx.x;
  const float* r = rois + roi * 5;

  int b  = (int)r[0];
  b = min(max(b, 0), B_ - 1);
  int x1 = (int)rintf(r[1] * SCALE);
  int y1 = (int)rintf(r[2] * SCALE);
  int x2 = (int)rintf(r[3] * SCALE);
  int y2 = (int)rintf(r[4] * SCALE);

  // Exact reference arithmetic: extent = max(hi-lo+1, 1) (int -> f32); bin = extent / 7
  float ew  = (float)max(x2 - x1 + 1, 1);
  float eh  = (float)max(y2 - y1 + 1, 1);
  float bwf = ew / 7.0f;
  float bhf = eh / 7.0f;

  // Tile = union of all bins (start(0) = x1, end(6) = ceil(7*bin)+x1), clipped to [0, size]
  int xs0 = min(max(x1, 0), W_);
  int xe0 = min(max((int)ceilf(7.0f * bwf) + x1, 0), W_);
  int ys0 = min(max(y1, 0), H_);
  int ye0 = min(max((int)ceilf(7.0f * bhf) + y1, 0), H_);
  int tw = xe0 - xs0;
  int th = ye0 - ys0;

  if (tw <= 0 || th <= 0) {  // whole RoI falls outside -> all bins empty -> zeros
    for (int i = tid; i < C_ * 49; i += 256)
      out[(size_t)roi * C_ * 49 + i] = 0.0f;
    return;
  }

  const int tile = tw * th;                 // <= 3800 < LDSF always
  const int cc   = min(C_, LDSF / tile);    // channels per chunk (>= 2)
  const u32 lds_base = (u32)(uintptr_t)(void*)&lds[0][0]; // low 32 bits == LDS offset
  const float* gchunk0 = feat + (((size_t)b * C_) * H_ + ys0) * W_ + xs0;

  const bool issuer = (tid < 32);           // wave 0 only (wave32)
  int buf = 0;

  if (issuer) {
    tdm_issue(gchunk0, lds_base, tw, th, min(cc, C_));
    __builtin_amdgcn_s_wait_tensorcnt(0);
  }
  __syncthreads();

  for (int c0 = 0; c0 < C_; c0 += cc) {
    const int cn = min(cc, C_ - c0);
    const int c1 = c0 + cc;

    // Prefetch next chunk into the other buffer while we compute this one.
    if (c1 < C_ && issuer)
      tdm_issue(gchunk0 + (size_t)c1 * H_ * W_,
                lds_base + (u32)(buf ^ 1) * (LDSF * 4),
                tw, th, min(cc, C_ - c1));

    const float* lbuf = &lds[buf][0];
    for (int i = tid; i < cn * 49; i += 256) {
      int cl = i / 49;
      int rb = i - cl * 49;
      int ph = rb / 7;
      int pw = rb - ph * 7;

      int ws = min(max((int)floorf((float)pw * bwf) + x1, 0), W_);
      int we = min(max((int)ceilf(((float)pw + 1.0f) * bwf) + x1, 0), W_);
      int hs = min(max((int)floorf((float)ph * bhf) + y1, 0), H_);
      int he = min(max((int)ceilf(((float)ph + 1.0f) * bhf) + y1, 0), H_);

      float m = -INFINITY;
      const float* cbase = lbuf + cl * tile - ys0 * tw - xs0;
      for (int y = hs; y < he; ++y) {
        const float* rp = cbase + y * tw;
        for (int x = ws; x < we; ++x) m = fmaxf(m, rp[x]);
      }
      out[(size_t)roi * C_ * 49 + (size_t)(c0 + cl) * 49 + rb] =
          __builtin_isfinite(m) ? m : 0.0f;
    }
    __syncthreads();                         // everyone done reading lds[buf]
    if (c1 < C_ && issuer)
      __builtin_amdgcn_s_wait_tensorcnt(0);  // next chunk's DMA complete
    __syncthreads();                         // publish lds[buf^1] to all waves
    buf ^= 1;
  }
}

extern "C" void kernel_launch(void* const* d_in, const int* in_sizes, int n_in,
                              void* d_out, int out_size, void* d_ws, size_t ws_size,
                              hipStream_t stream) {
  const float* feat = (const float*)d_in[0];  // (2, 256, 50, 76) f32
  const float* rois = (const float*)d_in[1];  // (128, 5) f32
  float* out = (float*)d_out;                 // (128, 256, 7, 7) f32
  roipool_tdm_kernel<<<NROIS, 256, 0, stream>>>(feat, rois, out);
}